// Encoder_19713899888647
// MI455X (gfx1250) — compile-verified
//
#include <hip/hip_runtime.h>

typedef __attribute__((ext_vector_type(2))) float v2f;
typedef __attribute__((ext_vector_type(8))) float v8f;
typedef __attribute__((ext_vector_type(4))) int   v4i;

#define STATE_DIM 256
#define HIDDEN    128
#define LATENT    64
#define NCODES    16384
#define BATCH     8192

// ---------------------------------------------------------------------------
// Async global->LDS copy (CDNA5 GLOBAL_LOAD_ASYNC_TO_LDS_B128, ASYNCcnt
// tracked). Builtin takes v4i* pointers (global AS1 src, LDS AS3 dst).
// ---------------------------------------------------------------------------
__device__ __forceinline__ void async_copy_b128(const float* src,
                                                float* dst_lds)
{
#if __has_builtin(__builtin_amdgcn_global_load_async_to_lds_b128)
    __builtin_amdgcn_global_load_async_to_lds_b128(
        (__attribute__((address_space(1))) v4i*)(v4i*)(uintptr_t)src,
        (__attribute__((address_space(3))) v4i*)(v4i*)(uintptr_t)dst_lds,
        0, 0);
#else
    *(float4*)dst_lds = *(const float4*)src;
#endif
}

__device__ __forceinline__ void wait_async_lds()
{
#if __has_builtin(__builtin_amdgcn_global_load_async_to_lds_b128)
#if __has_builtin(__builtin_amdgcn_s_wait_asynccnt)
    __builtin_amdgcn_s_wait_asynccnt(0);
#else
    asm volatile("s_wait_asynccnt 0x0" ::: "memory");
#endif
#endif
}

// ---------------------------------------------------------------------------
// Kernel 1: z_e = relu(x @ W1 + b1) @ W2 + b2
// 128 threads (4 waves), 64 rows per workgroup. f32 WMMA 16x16x4.
// A layout (16x4 f32): lanes 0-15 M=lane K={k0,k0+1}; lanes 16-31 K={k0+2,k0+3}
// B layout (4x16 f32): lanes 0-15 N=lane K={k0,k0+1}; lanes 16-31 K={k0+2,k0+3}
// D layout (16x16 f32): vgpr v -> M = v + 8*(lane>=16), N = lane%16
// ---------------------------------------------------------------------------
__global__ __launch_bounds__(128) void mlp_kernel(
    const float* __restrict__ x,  const float* __restrict__ W1,
    const float* __restrict__ b1, const float* __restrict__ W2,
    const float* __restrict__ b2, float* __restrict__ z_e)
{
    constexpr int ROWS = 64;
    constexpr int HPAD = HIDDEN + 4;           // padded LDS stride (bank spread)
    __shared__ float Hs[ROWS * HPAD];          // 64*132*4 = 33.8 KB

    const int lane  = threadIdx.x & 31;
    const int wave  = threadIdx.x >> 5;        // 0..3
    const int lrow  = lane & 15;
    const int khalf = lane >> 4;               // 0 or 1
    const int m0    = blockIdx.x * ROWS + wave * 16;

    // ---- Stage 1: H = relu(X @ W1 + b1) for rows m0..m0+15 ----
    for (int n0 = 0; n0 < HIDDEN; n0 += 16) {
        v8f acc = {};
        for (int k0 = 0; k0 < STATE_DIM; k0 += 4) {
            const float* ap = x + (size_t)(m0 + lrow) * STATE_DIM + k0 + 2 * khalf;
            v2f a; a.x = ap[0]; a.y = ap[1];
            const float* bp = W1 + (size_t)(k0 + 2 * khalf) * HIDDEN + n0 + lrow;
            v2f b; b.x = bp[0]; b.y = bp[HIDDEN];
            acc = __builtin_amdgcn_wmma_f32_16x16x4_f32(
                false, a, false, b, (short)0, acc, false, false);
        }
        const float bias = b1[n0 + lrow];
#pragma unroll
        for (int v = 0; v < 8; ++v) {
            float h = acc[v] + bias;
            h = h > 0.0f ? h : 0.0f;
            const int row = wave * 16 + v + 8 * khalf;
            Hs[row * HPAD + n0 + lrow] = h;
        }
    }
    __syncthreads();

    // ---- Stage 2: Z = H @ W2 + b2 ----
    for (int n0 = 0; n0 < LATENT; n0 += 16) {
        v8f acc = {};
        for (int k0 = 0; k0 < HIDDEN; k0 += 4) {
            const float* ap = &Hs[(wave * 16 + lrow) * HPAD + k0 + 2 * khalf];
            v2f a; a.x = ap[0]; a.y = ap[1];
            const float* bp = W2 + (size_t)(k0 + 2 * khalf) * LATENT + n0 + lrow;
            v2f b; b.x = bp[0]; b.y = bp[LATENT];
            acc = __builtin_amdgcn_wmma_f32_16x16x4_f32(
                false, a, false, b, (short)0, acc, false, false);
        }
        const float bias = b2[n0 + lrow];
#pragma unroll
        for (int v = 0; v < 8; ++v) {
            const int row = m0 + v + 8 * khalf;
            z_e[(size_t)row * LATENT + n0 + lrow] = acc[v] + bias;
        }
    }
}

// ---------------------------------------------------------------------------
// Kernel 2: cnorm[k] = 0.5 * ||codebook[k]||^2
// ---------------------------------------------------------------------------
__global__ void cnorm_kernel(const float* __restrict__ codebook,
                             float* __restrict__ cnorm)
{
    const int k = blockIdx.x * blockDim.x + threadIdx.x;
    if (k >= NCODES) return;
    const float4* cp = (const float4*)(codebook + (size_t)k * LATENT);
    float s = 0.0f;
#pragma unroll
    for (int j = 0; j < LATENT / 4; ++j) {
        const float4 c = cp[j];
        s += c.x * c.x + c.y * c.y + c.z * c.z + c.w * c.w;
    }
    cnorm[k] = 0.5f * s;
}

// ---------------------------------------------------------------------------
// Kernel 3: nearest-code search + gather.
// 256 threads (8 waves), 128 rows per WG. Per wave: 16 rows, K=64 A-fragments
// in registers; 32-code B tiles double-buffered in padded LDS, staged with
// async global->LDS copies. argmin_k( 0.5*||c_k||^2 - z.c_k ) == argmin d2.
// ---------------------------------------------------------------------------
__global__ __launch_bounds__(256) void vq_kernel(
    const float* __restrict__ z_e, const float* __restrict__ codebook,
    const float* __restrict__ cnorm, float* __restrict__ z_q,
    int* __restrict__ indices)
{
    constexpr int CTILE = 32;                  // codes per iteration
    constexpr int CPAD  = LATENT + 4;          // 68 floats: conflict-free banks
    constexpr int NTILE = NCODES / CTILE;      // 512
    __shared__ float Cs[2][CTILE * CPAD];      // 2 * 8.5 KB
    __shared__ float CNs[2][CTILE];

    const int lane  = threadIdx.x & 31;
    const int wave  = threadIdx.x >> 5;        // 0..7
    const int lrow  = lane & 15;
    const int khalf = lane >> 4;
    const int m0    = blockIdx.x * 128 + wave * 16;

    // Preload this wave's A fragments: 16 rows x K=64 -> 16 x v2f (32 VGPRs)
    v2f afrag[16];
#pragma unroll
    for (int ks = 0; ks < 16; ++ks) {
        const float* ap = z_e + (size_t)(m0 + lrow) * LATENT + ks * 4 + 2 * khalf;
        afrag[ks].x = ap[0];
        afrag[ks].y = ap[1];
    }

    // Cooperative tile load: 2048 floats = 32 codes x 64; 256 threads x 8 f32
    // (two async b128 copies each). Fully coalesced 32B per thread.
    auto load_tile = [&](int t, int buf) {
        const int i   = threadIdx.x;
        const int row = i >> 3;                // 0..31 (code within tile)
        const int col = (i & 7) * 8;           // 0..56
        const float* src = codebook + (size_t)(t * CTILE + row) * LATENT + col;
        float* dst = &Cs[buf][row * CPAD + col];
        async_copy_b128(src,     dst);
        async_copy_b128(src + 4, dst + 4);
        if (i < CTILE) CNs[buf][i] = cnorm[t * CTILE + i];
    };

    float best[2][8];
    int   bidx[2][8];
#pragma unroll
    for (int n = 0; n < 2; ++n)
#pragma unroll
        for (int v = 0; v < 8; ++v) { best[n][v] = 3.4e38f; bidx[n][v] = 0; }

    load_tile(0, 0);
    wait_async_lds();
    __syncthreads();

    for (int t = 0; t < NTILE; ++t) {
        const int cur = t & 1;
        // Prefetch next tile into the other buffer (its last readers finished
        // at the barrier that ended iteration t-1).
        if (t + 1 < NTILE) load_tile(t + 1, cur ^ 1);

        v8f acc0 = {};
        v8f acc1 = {};
#pragma unroll
        for (int ks = 0; ks < 16; ++ks) {
            // B frags: lane -> code {lrow, 16+lrow}, K = ks*4 + 2*khalf (+1)
            const float* b0p = &Cs[cur][lrow * CPAD + ks * 4 + 2 * khalf];
            const float* b1p = b0p + 16 * CPAD;
            v2f b0; b0.x = b0p[0]; b0.y = b0p[1];
            v2f b1; b1.x = b1p[0]; b1.y = b1p[1];
            acc0 = __builtin_amdgcn_wmma_f32_16x16x4_f32(
                false, afrag[ks], false, b0, (short)0, acc0, false, false);
            acc1 = __builtin_amdgcn_wmma_f32_16x16x4_f32(
                false, afrag[ks], false, b1, (short)0, acc1, false, false);
        }

        const float cn0 = CNs[cur][lrow];
        const float cn1 = CNs[cur][16 + lrow];
        const int   c0  = t * CTILE + lrow;
        const int   c1  = c0 + 16;
#pragma unroll
        for (int v = 0; v < 8; ++v) {
            const float d0 = cn0 - acc0[v];        // 0.5||c||^2 - z.c
            const float d1 = cn1 - acc1[v];
            if (d0 < best[0][v]) { best[0][v] = d0; bidx[0][v] = c0; }
            if (d1 < best[1][v]) { best[1][v] = d1; bidx[1][v] = c1; }
        }
        wait_async_lds();   // all of next tile landed in LDS (per-wave ASYNCcnt)
        __syncthreads();    // visible to all waves; current buffer reusable
    }

    // Merge the two column-halves, then reduce across the 16 lanes of each
    // half-wave. Row for vgpr v: m0 + v + 8*khalf.
#pragma unroll
    for (int v = 0; v < 8; ++v) {
        float bv = best[0][v];
        int   bi = bidx[0][v];
        if (best[1][v] < bv || (best[1][v] == bv && bidx[1][v] < bi)) {
            bv = best[1][v]; bi = bidx[1][v];
        }
#pragma unroll
        for (int off = 8; off >= 1; off >>= 1) {
            const float ov = __shfl_xor(bv, off, 16);
            const int   oi = __shfl_xor(bi, off, 16);
            if (ov < bv || (ov == bv && oi < bi)) { bv = ov; bi = oi; }
        }
        const int row = m0 + v + 8 * khalf;
        if (lrow == 0) indices[row] = bi;
        // Gather z_q row: 16 lanes x float4 = 64 floats (bi uniform per group)
        const float4 c = *(const float4*)(codebook + (size_t)bi * LATENT + lrow * 4);
        *(float4*)(z_q + (size_t)row * LATENT + lrow * 4) = c;
    }
}

// ---------------------------------------------------------------------------
extern "C" void kernel_launch(void* const* d_in, const int* in_sizes, int n_in,
                              void* d_out, int out_size, void* d_ws, size_t ws_size,
                              hipStream_t stream)
{
    const float* x        = (const float*)d_in[0];
    const float* W1       = (const float*)d_in[1];
    const float* b1       = (const float*)d_in[2];
    const float* W2       = (const float*)d_in[3];
    const float* b2       = (const float*)d_in[4];
    const float* codebook = (const float*)d_in[5];

    float* z_e   = (float*)d_ws;                        // 8192*64 f32 = 2 MB
    float* cnorm = z_e + (size_t)BATCH * LATENT;        // 16384 f32

    float* z_q     = (float*)d_out;                     // output 0: [8192,64] f32
    int*   indices = (int*)((float*)d_out + (size_t)BATCH * LATENT); // output 1

    mlp_kernel<<<BATCH / 64, 128, 0, stream>>>(x, W1, b1, W2, b2, z_e);
    cnorm_kernel<<<NCODES / 256, 256, 0, stream>>>(codebook, cnorm);
    vq_kernel<<<BATCH / 128, 256, 0, stream>>>(z_e, codebook, cnorm, z_q, indices);
}